// QuantumHydraLayer_19112604467495
// MI455X (gfx1250) — compile-verified
//
#include <hip/hip_runtime.h>
#include <math.h>

typedef __attribute__((ext_vector_type(2))) float v2f;
typedef __attribute__((ext_vector_type(8))) float v8f;

#define NQ      12
#define DIM     4096          // 2^12
#define HALF    2048
#define NGATE   96            // N_QLCU = 4*12*2
#define NQD     36
#define TWOPI_F 6.28318f      // 2 * 3.14159 (matches reference literal)

// ---------------------------------------------------------------------------
// Kernel 1: qp = sigmoid(x @ W_proj^T + b_proj) * 2pi + qlcu_base
// M=512, N=96, K=128  ->  32 x 6 tiles of 16x16, K-loop of 32 x (k=4) WMMA ops.
// One wave per tile, 4 waves per block, 48 blocks.
// ---------------------------------------------------------------------------
__global__ __launch_bounds__(128) void proj_wmma_kernel(
    const float* __restrict__ x,       // [512,128]
    const float* __restrict__ W,       // [96,128]
    const float* __restrict__ bproj,   // [96]
    const float* __restrict__ qbase,   // [96]
    float* __restrict__ qp)            // [512,96]
{
    const int lane = threadIdx.x & 31;
    const int wave = threadIdx.x >> 5;
    const int tile = blockIdx.x * 4 + wave;   // 0..191
    const int tM = tile / 6;                  // 0..31
    const int tN = tile % 6;                  // 0..5
    const int l16  = lane & 15;
    const int klo  = (lane >> 4) * 2;         // lanes 0-15 -> K+0/1, 16-31 -> K+2/3
    const int mRow = tM * 16 + l16;           // A: M = lane%16
    const int nRow = tN * 16 + l16;           // B: N = lane%16 (B = W^T, row-of-W)

    v8f acc = (v8f)0.0f;
    const float* xr = x + mRow * 128;
    const float* wr = W + nRow * 128;
    #pragma unroll 4
    for (int k0 = 0; k0 < 128; k0 += 4) {
        v2f a, b;
        a.x = xr[k0 + klo];
        a.y = xr[k0 + klo + 1];
        b.x = wr[k0 + klo];
        b.y = wr[k0 + klo + 1];
        // D = A(16x4) * B(4x16) + C   -> v_wmma_f32_16x16x4_f32
        acc = __builtin_amdgcn_wmma_f32_16x16x4_f32(
            false, a, false, b, (short)0, acc, false, false);
    }

    // C/D layout: VGPR r -> M = tM*16 + r (+8 for lanes 16-31), N = lane%16
    const int colN  = tN * 16 + l16;
    const float bp  = bproj[colN];
    const float qb  = qbase[colN];
    const int mBase = tM * 16 + ((lane & 16) ? 8 : 0);
    #pragma unroll
    for (int r = 0; r < 8; ++r) {
        float v = acc[r] + bp;
        float sgm = 1.0f / (1.0f + expf(-v));
        qp[(mBase + r) * NGATE + colN] = sgm * TWOPI_F + qb;
    }
}

// ---------------------------------------------------------------------------
// LDS gate helpers (state = 4096 complex floats, 256 threads, 8 pairs/thread).
// Callers unroll the wire loop so k / ka / kb are compile-time constants:
// all shift amounts and pair offsets constant-fold.
// Wire i acts on flat-index bit k = NQ-1-i (axis 1 is MSB).
// ---------------------------------------------------------------------------
__device__ __forceinline__ void gate_ry(float2* s, int k, float c, float sn, int tid)
{
    #pragma unroll 4
    for (int j = 0; j < HALF / 256; ++j) {
        int p  = tid + j * 256;
        int i0 = ((p >> k) << (k + 1)) | (p & ((1 << k) - 1));
        int i1 = i0 | (1 << k);
        float2 a0 = s[i0], a1 = s[i1];
        s[i0] = make_float2(c * a0.x - sn * a1.x, c * a0.y - sn * a1.y);
        s[i1] = make_float2(sn * a0.x + c * a1.x, sn * a0.y + c * a1.y);
    }
    __syncthreads();
}

// exp(-i th/2 P(x)P): out[idx] = c*in[idx] - i*(s*sgn)*in[idx^mask]
// XX: sgn=+1.  YY: sgn shared per pair = (bit_kb(idx0)==0 ? -1 : +1), idx0 has bit_ka=0.
__device__ __forceinline__ void gate_ising(float2* s, int ka, int kb,
                                           float c, float sn, bool yy, int tid)
{
    const int mask = (1 << ka) | (1 << kb);
    #pragma unroll 4
    for (int j = 0; j < HALF / 256; ++j) {
        int p  = tid + j * 256;
        int i0 = ((p >> ka) << (ka + 1)) | (p & ((1 << ka) - 1));  // bit ka == 0
        int i1 = i0 ^ mask;
        float ss = sn;
        if (yy) ss = (((i0 >> kb) & 1) ? sn : -sn);
        float2 a0 = s[i0], a1 = s[i1];
        // c*a - i*ss*b  ->  (c*ax + ss*by, c*ay - ss*bx)
        s[i0] = make_float2(c * a0.x + ss * a1.y, c * a0.y - ss * a1.x);
        s[i1] = make_float2(c * a1.x + ss * a0.y, c * a1.y - ss * a0.x);
    }
    __syncthreads();
}

// ---------------------------------------------------------------------------
// Kernel 2: psi3 = qd(|0>)  — batch independent, one workgroup.
// ---------------------------------------------------------------------------
__global__ __launch_bounds__(256) void qd_kernel(const float* __restrict__ qd,
                                                 float2* __restrict__ psi3g)
{
    __shared__ float2 s[DIM];
    __shared__ float gc[NQD], gs[NQD];
    const int tid = threadIdx.x;
    for (int j = tid; j < DIM; j += 256) s[j] = make_float2(0.f, 0.f);
    if (tid < NQD) {                       // per-gate coefficients, once per block
        float th = 0.5f * qd[tid];
        gc[tid] = cosf(th);
        gs[tid] = sinf(th);
    }
    __syncthreads();
    if (tid == 0) s[0] = make_float2(1.f, 0.f);
    __syncthreads();

    // RX(qd[i]) : n0 = c*a0 - i*s*a1 ; n1 = -i*s*a0 + c*a1
    #pragma unroll
    for (int i = 0; i < NQ; ++i) {
        const int k = NQ - 1 - i;
        const float c = gc[i], sn = gs[i];
        #pragma unroll 4
        for (int j = 0; j < HALF / 256; ++j) {
            int p  = tid + j * 256;
            int i0 = ((p >> k) << (k + 1)) | (p & ((1 << k) - 1));
            int i1 = i0 | (1 << k);
            float2 a0 = s[i0], a1 = s[i1];
            s[i0] = make_float2(c * a0.x + sn * a1.y, c * a0.y - sn * a1.x);
            s[i1] = make_float2(sn * a0.y + c * a1.x, -sn * a0.x + c * a1.y);
        }
        __syncthreads();
    }
    // RY(qd[12+i])
    #pragma unroll
    for (int i = 0; i < NQ; ++i)
        gate_ry(s, NQ - 1 - i, gc[NQ + i], gs[NQ + i], tid);
    // RZ(qd[24+i]) : a0 *= e^{-i th/2} ; a1 *= e^{+i th/2}
    #pragma unroll
    for (int i = 0; i < NQ; ++i) {
        const int k = NQ - 1 - i;
        const float c = gc[2 * NQ + i], sn = gs[2 * NQ + i];
        #pragma unroll 4
        for (int j = 0; j < HALF / 256; ++j) {
            int p  = tid + j * 256;
            int i0 = ((p >> k) << (k + 1)) | (p & ((1 << k) - 1));
            int i1 = i0 | (1 << k);
            float2 a0 = s[i0], a1 = s[i1];
            s[i0] = make_float2(c * a0.x + sn * a0.y, c * a0.y - sn * a0.x);
            s[i1] = make_float2(c * a1.x - sn * a1.y, c * a1.y + sn * a1.x);
        }
        __syncthreads();
    }
    for (int j = tid; j < DIM; j += 256) psi3g[j] = s[j];
}

// ---------------------------------------------------------------------------
// Kernel 3: per-batch QLCU + combine (index permutations replace shift/flip)
// + normalize + measure + output projection.
// ---------------------------------------------------------------------------
__global__ __launch_bounds__(256) void hydra_kernel(
    const float*  __restrict__ qp,     // [512,96]
    const float2* __restrict__ psi3,   // [4096]
    const float*  __restrict__ alpha,  // [2]
    const float*  __restrict__ beta,   // [2]
    const float*  __restrict__ gamma,  // [2]
    const float*  __restrict__ Wout,   // [1,36]
    const float*  __restrict__ bout,   // [1]
    float* __restrict__ out)           // [512]
{
    __shared__ float2 s[DIM];
    __shared__ float gc[NGATE], gs[NGATE];
    __shared__ float red[37];          // [0..35] measurements, [36] squared norm
    const int tid = threadIdx.x;
    const int b   = blockIdx.x;
    const float* qpb = qp + b * NGATE;

    // warm psi3 (one 128B cacheline per thread) while the QLCU runs
    __builtin_prefetch((const char*)psi3 + tid * 128, 0, 1);

    for (int j = tid; j < DIM; j += 256) s[j] = make_float2(0.f, 0.f);
    if (tid < NGATE) {                 // per-gate coefficients, once per block
        float th = 0.5f * qpb[tid];
        gc[tid] = cosf(th);
        gs[tid] = sinf(th);
    }
    if (tid < 37) red[tid] = 0.f;
    __syncthreads();
    if (tid == 0) s[0] = make_float2(1.f, 0.f);
    __syncthreads();

    // ---- QLCU: Q = qlcu(|0>) ; (qlcu(flip(|0>)) == Q since flip(|0>)==|0>) ----
    for (int layer = 0; layer < 2; ++layer) {
        const int base = layer * 48;
        #pragma unroll
        for (int i = 0; i < NQ; ++i)                          // RY sweep
            gate_ry(s, NQ - 1 - i, gc[base + i], gs[base + i], tid);
        #pragma unroll
        for (int i = 0; i < NQ; ++i)                          // XX sweep
            gate_ising(s, NQ - 1 - i, NQ - 1 - ((i + 1) % NQ),
                       gc[base + 12 + i], gs[base + 12 + i], false, tid);
        #pragma unroll
        for (int i = 0; i < NQ; ++i)                          // RY sweep
            gate_ry(s, NQ - 1 - i, gc[base + 24 + i], gs[base + 24 + i], tid);
        #pragma unroll
        for (int i = 0; i < NQ; ++i)                          // YY sweep
            gate_ising(s, NQ - 1 - i, NQ - 1 - ((i + 1) % NQ),
                       gc[base + 36 + i], gs[base + 36 + i], true, tid);
    }

    // ---- combine: comb[g] = a*Q[rotr(g)] + b*Q[rotr(bitrev(g))] + g*psi3[g] ----
    const float aR = alpha[0], aI = alpha[1];
    const float bR = beta[0],  bI = beta[1];
    const float gR = gamma[0], gI = gamma[1];
    float2 comb[DIM / 256];
    float nrm = 0.f;
    #pragma unroll
    for (int j = 0; j < DIM / 256; ++j) {
        int g  = tid + j * 256;
        int g1 = ((g >> 1) | ((g & 1) << (NQ - 1))) & (DIM - 1);   // rotr12
        int gr = (int)(__brev((unsigned)g) >> (32 - NQ));          // bitrev12
        int g2 = ((gr >> 1) | ((gr & 1) << (NQ - 1))) & (DIM - 1);
        float2 q1 = s[g1], q2 = s[g2], p3 = psi3[g];
        float cr = aR * q1.x - aI * q1.y + bR * q2.x - bI * q2.y + gR * p3.x - gI * p3.y;
        float ci = aR * q1.y + aI * q1.x + bR * q2.y + bI * q2.x + gR * p3.y + gI * p3.x;
        comb[j] = make_float2(cr, ci);
        nrm += cr * cr + ci * ci;
    }
    atomicAdd(&red[36], nrm);
    __syncthreads();                                // all reads of s + norm done
    const float inv = 1.0f / (sqrtf(red[36]) + 1e-9f);
    #pragma unroll
    for (int j = 0; j < DIM / 256; ++j) {
        int g = tid + j * 256;
        s[g] = make_float2(comb[j].x * inv, comb[j].y * inv);
    }
    __syncthreads();

    // ---- measure <X_i>, <Y_i>, <Z_i> ----
    #pragma unroll
    for (int i = 0; i < NQ; ++i) {
        const int k = NQ - 1 - i;
        float sx = 0.f, sy = 0.f, sz = 0.f;
        #pragma unroll 4
        for (int j = 0; j < HALF / 256; ++j) {
            int p  = tid + j * 256;
            int i0 = ((p >> k) << (k + 1)) | (p & ((1 << k) - 1));
            int i1 = i0 | (1 << k);
            float2 p0 = s[i0], p1 = s[i1];
            sx += p0.x * p1.x + p0.y * p1.y;         // Re(conj(p0)*p1)
            sy += p0.x * p1.y - p0.y * p1.x;         // Im(conj(p0)*p1)
            sz += p0.x * p0.x + p0.y * p0.y - p1.x * p1.x - p1.y * p1.y;
        }
        atomicAdd(&red[i],          2.f * sx);
        atomicAdd(&red[NQ + i],     2.f * sy);
        atomicAdd(&red[2 * NQ + i], sz);
    }
    __syncthreads();

    if (tid == 0) {
        float o = bout[0];
        #pragma unroll
        for (int m = 0; m < 3 * NQ; ++m) o += red[m] * Wout[m];
        out[b] = o;
    }
}

// ---------------------------------------------------------------------------
extern "C" void kernel_launch(void* const* d_in, const int* in_sizes, int n_in,
                              void* d_out, int out_size, void* d_ws, size_t ws_size,
                              hipStream_t stream) {
    const float* x      = (const float*)d_in[0];   // [512,128]
    const float* W_proj = (const float*)d_in[1];   // [96,128]
    const float* b_proj = (const float*)d_in[2];   // [96]
    const float* q_base = (const float*)d_in[3];   // [96]
    const float* qd_p   = (const float*)d_in[4];   // [36]
    const float* alpha  = (const float*)d_in[5];   // [2]
    const float* beta   = (const float*)d_in[6];   // [2]
    const float* gamma  = (const float*)d_in[7];   // [2]
    const float* W_out  = (const float*)d_in[8];   // [1,36]
    const float* b_out  = (const float*)d_in[9];   // [1]
    float* out = (float*)d_out;                    // [512]

    // workspace layout: qp [512*96 f32] then psi3 [4096 float2]
    float*  ws_qp   = (float*)d_ws;
    float2* ws_psi3 = (float2*)((char*)d_ws + 512 * NGATE * sizeof(float));

    proj_wmma_kernel<<<48, 128, 0, stream>>>(x, W_proj, b_proj, q_base, ws_qp);
    qd_kernel<<<1, 256, 0, stream>>>(qd_p, ws_psi3);
    hydra_kernel<<<512, 256, 0, stream>>>(ws_qp, ws_psi3, alpha, beta, gamma,
                                          W_out, b_out, out);
}